// HSTUPositionalEncoder_40080634806844
// MI455X (gfx1250) — compile-verified
//
#include <hip/hip_runtime.h>
#include <cmath>

// HSTU positional encoder: out[t,:] = seq_emb[t,:]*sqrt(D)
//                                    + pos_emb[pos(t),:] + ts_emb[bucket(t),:]
// Memory-bound (AI ~= 0.26 flop/byte); roofline floor ~11us at 23.3 TB/s for
// the 256MB of unavoidable stream traffic. MI455X strategy:
//  - float4 (b128) coalesced loads/stores; 128 lanes cover one 2KB row
//  - NT hints on the 256MB read-once/write-once streams so the 20MB gather
//    tables stay resident in the 192MB L2
//  - gfx1250 global_prefetch_b8 for the next iteration's gathered rows
//  - per-token index math once per token into LDS (16 tokens/block)
//  - guard-free fully-unrolled hot path for full tiles (T % 16 == 0)

typedef float f32x4 __attribute__((ext_vector_type(4)));

constexpr int TOK_PER_BLOCK = 16;
constexpr int THREADS       = 256;   // 8 wave32 waves

__global__ __launch_bounds__(THREADS)
void hstu_pos_enc_d512(const int*   __restrict__ seq_lengths,
                       const int*   __restrict__ seq_offsets,   // B+1 entries
                       const float* __restrict__ seq_emb,       // T x D
                       const int*   __restrict__ num_targets,
                       const int*   __restrict__ seq_ts,        // T
                       const float* __restrict__ pos_emb,       // NPOS x D
                       const float* __restrict__ ts_emb,        // (NTIME+1) x D
                       float*       __restrict__ out,           // T x D
                       int B, int T, int NPOS, int NTIME, float alpha)
{
    constexpr int D    = 512;
    constexpr int CH   = D / 4;             // 128 float4 chunks per row
    constexpr int SUBS = THREADS / CH;      // 2 rows in flight per iteration
    constexpr int ITER = TOK_PER_BLOCK / SUBS;  // 8

    __shared__ int s_pos[TOK_PER_BLOCK];
    __shared__ int s_bkt[TOK_PER_BLOCK];

    const int tid  = threadIdx.x;
    const int tok0 = blockIdx.x * TOK_PER_BLOCK;

    // ---- phase 1: 16 threads compute per-token gather indices ----
    if (tid < TOK_PER_BLOCK) {
        int t = tok0 + tid;
        int pos = 0, bkt = 0;
        if (t < T) {
            // largest s in [0,B-1] with seq_offsets[s] <= t  (searchsorted right - 1)
            int lo = 0, hi = B - 1;
            while (lo < hi) {
                int mid = (lo + hi + 1) >> 1;
                if (seq_offsets[mid] <= t) lo = mid; else hi = mid - 1;
            }
            const int s     = lo;
            const int off   = seq_offsets[s];
            const int L     = seq_lengths[s];
            const int local = t - off;

            int high = L - num_targets[s];          // interleave_targets=False
            if (high > NPOS - 1) high = NPOS - 1;
            pos = local < high ? local : high;
            if (pos < 0) pos = 0;
            if (pos > NPOS - 1) pos = NPOS - 1;

            // time bucket: floor(sqrt(f32(delta))) — same f32 path as reference
            long long q  = (long long)seq_ts[off + L - 1];
            long long dt = q - (long long)seq_ts[t];
            if (dt < 0) dt = 0;
            int b = (int)floorf(sqrtf((float)dt));
            if (b < 0) b = 0;
            if (b > NTIME) b = NTIME;
            bkt = b;

            // warm the first lines of this token's gather rows (global_prefetch_b8)
            __builtin_prefetch(pos_emb + (size_t)pos * D, 0, 3);
            __builtin_prefetch(ts_emb  + (size_t)bkt * D, 0, 3);
        }
        s_pos[tid] = pos;
        s_bkt[tid] = bkt;
    }
    __syncthreads();

    // ---- phase 2: stream 16 rows with b128 vector ops ----
    const int lane_c = tid & (CH - 1);     // float4 index within row: 0..127
    const int sub0   = tid / CH;           // 0..SUBS-1

    if (tok0 + TOK_PER_BLOCK <= T) {
        // Full tile: no bounds checks, compile-time trip count -> full unroll,
        // lets the scheduler clause/batch loads across iterations.
        int rpos[ITER], rbkt[ITER];
        #pragma unroll
        for (int k = 0; k < ITER; ++k) {
            const int ti = sub0 + k * SUBS;
            rpos[k] = s_pos[ti];
            rbkt[k] = s_bkt[ti];
        }
        #pragma unroll
        for (int k = 0; k < ITER; ++k) {
            const int ti = sub0 + k * SUBS;
            const int t  = tok0 + ti;

            // prefetch exactly the slices we will gather next iteration
            if (k + 1 < ITER) {
                __builtin_prefetch(pos_emb + (size_t)rpos[k + 1] * D + lane_c * 4, 0, 3);
                __builtin_prefetch(ts_emb  + (size_t)rbkt[k + 1] * D + lane_c * 4, 0, 3);
            }

            const f32x4* se = (const f32x4*)(seq_emb + (size_t)t       * D);
            const f32x4* pe = (const f32x4*)(pos_emb + (size_t)rpos[k] * D);
            const f32x4* te = (const f32x4*)(ts_emb  + (size_t)rbkt[k] * D);

            f32x4 a = __builtin_nontemporal_load(se + lane_c);  // read-once: NT
            f32x4 p = pe[lane_c];                               // L2 gather: RT
            f32x4 x = te[lane_c];                               // L2 gather: RT

            f32x4 r = a * alpha + p + x;

            __builtin_nontemporal_store(r, (f32x4*)(out + (size_t)t * D) + lane_c);
        }
    } else {
        // Ragged tail tile (not hit when T % TOK_PER_BLOCK == 0)
        for (int ti = sub0; ti < TOK_PER_BLOCK; ti += SUBS) {
            const int t = tok0 + ti;
            if (t >= T) break;
            const f32x4* se = (const f32x4*)(seq_emb + (size_t)t          * D);
            const f32x4* pe = (const f32x4*)(pos_emb + (size_t)s_pos[ti]  * D);
            const f32x4* te = (const f32x4*)(ts_emb  + (size_t)s_bkt[ti]  * D);
            f32x4 a = __builtin_nontemporal_load(se + lane_c);
            f32x4 r = a * alpha + pe[lane_c] + te[lane_c];
            __builtin_nontemporal_store(r, (f32x4*)(out + (size_t)t * D) + lane_c);
        }
    }
}

// Generic fallback for D != 512 (not expected with this harness; compile-safe).
__global__ __launch_bounds__(THREADS)
void hstu_pos_enc_generic(const int*   __restrict__ seq_lengths,
                          const int*   __restrict__ seq_offsets,
                          const float* __restrict__ seq_emb,
                          const int*   __restrict__ num_targets,
                          const int*   __restrict__ seq_ts,
                          const float* __restrict__ pos_emb,
                          const float* __restrict__ ts_emb,
                          float*       __restrict__ out,
                          int B, int T, int D, int NPOS, int NTIME, float alpha)
{
    __shared__ int s_pos, s_bkt;
    const int t = blockIdx.x;
    if (t >= T) return;
    if (threadIdx.x == 0) {
        int lo = 0, hi = B - 1;
        while (lo < hi) {
            int mid = (lo + hi + 1) >> 1;
            if (seq_offsets[mid] <= t) lo = mid; else hi = mid - 1;
        }
        const int s = lo, off = seq_offsets[s], L = seq_lengths[s];
        int high = L - num_targets[s];
        if (high > NPOS - 1) high = NPOS - 1;
        int pos = (t - off) < high ? (t - off) : high;
        if (pos < 0) pos = 0;
        if (pos > NPOS - 1) pos = NPOS - 1;
        long long dt = (long long)seq_ts[off + L - 1] - (long long)seq_ts[t];
        if (dt < 0) dt = 0;
        int b = (int)floorf(sqrtf((float)dt));
        if (b < 0) b = 0;
        if (b > NTIME) b = NTIME;
        s_pos = pos; s_bkt = b;
    }
    __syncthreads();
    const float* pe = pos_emb + (size_t)s_pos * D;
    const float* te = ts_emb  + (size_t)s_bkt * D;
    const float* se = seq_emb + (size_t)t * D;
    float*       ou = out     + (size_t)t * D;
    for (int d = threadIdx.x; d < D; d += THREADS)
        ou[d] = se[d] * alpha + pe[d] + te[d];
}

extern "C" void kernel_launch(void* const* d_in, const int* in_sizes, int n_in,
                              void* d_out, int out_size, void* d_ws, size_t ws_size,
                              hipStream_t stream) {
    // setup_inputs order:
    // 0 max_seq_len(1) 1 seq_lengths(B) 2 seq_offsets(B+1) 3 seq_embeddings(T*D)
    // 4 num_targets(B) 5 seq_timestamps(T) 6 pos_emb(NPOS*D) 7 ts_emb((NTIME+1)*D)
    const int*   seq_lengths = (const int*)  d_in[1];
    const int*   seq_offsets = (const int*)  d_in[2];
    const float* seq_emb     = (const float*)d_in[3];
    const int*   num_targets = (const int*)  d_in[4];
    const int*   seq_ts      = (const int*)  d_in[5];
    const float* pos_emb     = (const float*)d_in[6];
    const float* ts_emb      = (const float*)d_in[7];
    float*       out         = (float*)d_out;

    const int B     = in_sizes[1];
    const int T     = in_sizes[5];
    const int D     = (T > 0) ? in_sizes[3] / T : 0;
    const int NPOS  = (D > 0) ? in_sizes[6] / D : 1;
    const int NTIME = (D > 0) ? in_sizes[7] / D - 1 : 0;
    const float alpha = sqrtf((float)D);

    if (D == 512) {
        const int blocks = (T + TOK_PER_BLOCK - 1) / TOK_PER_BLOCK;
        hipLaunchKernelGGL(hstu_pos_enc_d512, dim3(blocks), dim3(THREADS), 0, stream,
                           seq_lengths, seq_offsets, seq_emb, num_targets, seq_ts,
                           pos_emb, ts_emb, out, B, T, NPOS, NTIME, alpha);
    } else {
        hipLaunchKernelGGL(hstu_pos_enc_generic, dim3(T > 0 ? T : 1), dim3(THREADS), 0, stream,
                           seq_lengths, seq_offsets, seq_emb, num_targets, seq_ts,
                           pos_emb, ts_emb, out, B, T, D, NPOS, NTIME, alpha);
    }
}